// Layer_2_4_9294309229010
// MI455X (gfx1250) — compile-verified
//
#include <hip/hip_runtime.h>

typedef __attribute__((ext_vector_type(32))) _Float16 v32h;
typedef __attribute__((ext_vector_type(16))) _Float16 v16h;
typedef __attribute__((ext_vector_type(8)))  _Float16 v8h;
typedef __attribute__((ext_vector_type(4)))  _Float16 v4h;
typedef __attribute__((ext_vector_type(8)))  float    v8f;
typedef __attribute__((ext_vector_type(4)))  float    v4f;

constexpr int Mdim = 256;
constexpr int Kdim = 8192;
constexpr int Ndim = 8192;

constexpr int MT = 64;            // M tile per workgroup
constexpr int NT = 128;           // N tile per workgroup (8 waves x 16 cols)
constexpr int KS = 64;            // K per step (SWMMAC K)
constexpr int LDSROW = 72;        // halves per LDS row (144B: 16B aligned, conflict-free)
constexpr int BUFB = MT * LDSROW * 2;   // 9216 bytes per tile buffer

// ---------------- one-time A fp32 -> f16 conversion into d_ws ----------------
__global__ __launch_bounds__(256)
void convA(const float* __restrict__ A, _Float16* __restrict__ A2)
{
    const int i = (blockIdx.x * 256 + threadIdx.x) * 8;
    v4f a0 = *(const v4f*)(A + i);
    v4f a1 = *(const v4f*)(A + i + 4);
    v8h hv;
    hv[0] = (_Float16)a0[0]; hv[1] = (_Float16)a0[1];
    hv[2] = (_Float16)a0[2]; hv[3] = (_Float16)a0[3];
    hv[4] = (_Float16)a1[0]; hv[5] = (_Float16)a1[1];
    hv[6] = (_Float16)a1[2]; hv[7] = (_Float16)a1[3];
    *(v8h*)(A2 + i) = hv;
}

// ---------------- main SWMMAC kernel: async-LDS double-buffered ----------------
__global__ __launch_bounds__(256)
void kern_async(const _Float16* __restrict__ A2,
                const int*      __restrict__ qvals,
                const int*      __restrict__ idx,
                const float*    __restrict__ s,
                float*          __restrict__ out)
{
    __shared__ __align__(16) unsigned char smem[2 * BUFB];   // 18432 B

    const int tid  = threadIdx.x;
    const int wave = tid >> 5;
    const int lane = tid & 31;
    const int ln16 = lane & 15;
    const int h    = lane >> 4;

    const int m0 = blockIdx.x * MT;                 // grid.x = m-tiles (fast dim)
    const int n0 = blockIdx.y * NT + wave * 16;
    const int n  = n0 + ln16;

    const unsigned ldsBase = (unsigned)(unsigned long long)(void*)smem;

    // staging: thread covers 32B of one 128B tile row, as 2 async 16B copies
    const int sr = tid >> 2;            // row 0..63
    const int sg = (tid & 3) * 32;      // byte offset within row
    const _Float16* gRow = A2 + (size_t)(m0 + sr) * Kdim + (sg >> 1);

    auto stage = [&](int kk, int buf) {
        unsigned long long g0 = (unsigned long long)(gRow + kk);
        unsigned           l0 = ldsBase + buf * BUFB + sr * 144 + sg;
        asm volatile("global_load_async_to_lds_b128 %0, %1, off"
                     :: "v"(l0), "v"(g0) : "memory");
        asm volatile("global_load_async_to_lds_b128 %0, %1, off"
                     :: "v"((unsigned)(l0 + 16)), "v"((unsigned long long)(g0 + 16))
                     : "memory");
    };

    v8f acc[4] = {};

    stage(0, 0);   // prologue: tile 0 in flight

    for (int k0 = 0; k0 < Kdim; k0 += KS) {
        const int cur = (k0 >> 6) & 1;

        // ---- decode sparse-A (weights): qvals rows ARE the SWMMAC packed layout
        const float sc    = s[(k0 >> 7) * Ndim + n];
        const float nsc8  = -8.0f * sc;              // fold zero-point into FMA
        const int   krow  = k0 >> 1;
        v16h aw;
        #pragma unroll
        for (int e = 0; e < 8; ++e) {
            const int q = qvals[(krow + h * 8 + e) * Ndim + n];
            aw[e] = (_Float16)__builtin_fmaf((float)q, sc, nsc8);
        }
        #pragma unroll
        for (int e = 0; e < 8; ++e) {
            const int q = qvals[(krow + 16 + h * 8 + e) * Ndim + n];
            aw[8 + e] = (_Float16)__builtin_fmaf((float)q, sc, nsc8);
        }

        // ---- 2-bit index register: 8 groups-of-4 per lane, 4 bits each
        unsigned ipack = 0;
        const int gg0 = (k0 >> 2) + h * 8;
        #pragma unroll
        for (int j = 0; j < 8; ++j) {
            const int2 ii = *(const int2*)(idx + ((size_t)(gg0 + j) * Ndim + n) * 2);
            ipack |= (unsigned)((ii.x & 3) | ((ii.y & 3) << 2)) << (4 * j);
        }

        __syncthreads();                  // B1: all waves done reading buf[cur^1]
        stage((k0 + KS) % Kdim, cur ^ 1); // prefetch next tile (wrapped dummy on last)
#if __has_builtin(__builtin_amdgcn_s_wait_asynccnt)
        __builtin_amdgcn_s_wait_asynccnt(2);   // oldest 2 (= buf[cur]) landed
#else
        asm volatile("s_wait_asynccnt 0x2" ::: "memory");
#endif
        __syncthreads();                  // B2: buf[cur] visible to all waves

        // ---- 4 SWMMACs along m, reusing the decoded sparse fragment
        const _Float16* As = (const _Float16*)(smem + cur * BUFB);
        #pragma unroll
        for (int mt = 0; mt < 4; ++mt) {
            const _Float16* rp = As + (mt * 16 + ln16) * LDSROW + h * 16;
            v16h blo = *(const v16h*)(rp);
            v16h bhi = *(const v16h*)(rp + 32);
            v32h bf  = __builtin_shufflevector(blo, bhi,
                0, 1, 2, 3, 4, 5, 6, 7, 8, 9, 10, 11, 12, 13, 14, 15,
                16, 17, 18, 19, 20, 21, 22, 23, 24, 25, 26, 27, 28, 29, 30, 31);
            acc[mt] = __builtin_amdgcn_swmmac_f32_16x16x64_f16(
                false, aw, false, bf, acc[mt], (int)ipack, false, false);
        }
    }

    __syncthreads();   // all compute done before reusing LDS for transpose

    // epilogue in buf1 (last dummy stage wrote buf0): transpose C^T -> coalesced
    float* Tw = (float*)(smem + BUFB) + wave * (16 * 17);
    #pragma unroll
    for (int mt = 0; mt < 4; ++mt) {
        #pragma unroll
        for (int r = 0; r < 8; ++r)                 // D[n = r+8h][m = ln16]
            Tw[(r + 8 * h) * 17 + ln16] = acc[mt][r];
        #pragma unroll
        for (int r = 0; r < 8; ++r)
            out[(size_t)(m0 + mt * 16 + r + 8 * h) * Ndim + n] = Tw[ln16 * 17 + (r + 8 * h)];
    }
}

// ---------------- fallback: fp32-staged SWMMAC (no workspace needed) ----------------
__global__ __launch_bounds__(256)
void kern_sync(const float* __restrict__ A,
               const int*   __restrict__ qvals,
               const int*   __restrict__ idx,
               const float* __restrict__ s,
               float*       __restrict__ out)
{
    __shared__ __align__(16) unsigned char smem[MT * LDSROW * 2];
    _Float16* As = (_Float16*)smem;

    const int tid  = threadIdx.x;
    const int wave = tid >> 5;
    const int lane = tid & 31;
    const int ln16 = lane & 15;
    const int h    = lane >> 4;

    const int m0 = blockIdx.x * MT;
    const int n0 = blockIdx.y * NT + wave * 16;
    const int n  = n0 + ln16;

    const int srow = tid >> 2;
    const int scol = (tid & 3) * 16;

    v8f acc[4] = {};

    for (int k0 = 0; k0 < Kdim; k0 += KS) {
        #pragma unroll
        for (int i = 0; i < 4; ++i) {
            v4f a4 = *(const v4f*)(A + (size_t)(m0 + srow) * Kdim + (k0 + scol + i * 4));
            v4h h4;
            h4[0] = (_Float16)a4[0]; h4[1] = (_Float16)a4[1];
            h4[2] = (_Float16)a4[2]; h4[3] = (_Float16)a4[3];
            *(v4h*)(As + srow * LDSROW + scol + i * 4) = h4;
        }

        const float sc   = s[(k0 >> 7) * Ndim + n];
        const float nsc8 = -8.0f * sc;
        const int   krow = k0 >> 1;
        v16h aw;
        #pragma unroll
        for (int e = 0; e < 8; ++e) {
            const int q = qvals[(krow + h * 8 + e) * Ndim + n];
            aw[e] = (_Float16)__builtin_fmaf((float)q, sc, nsc8);
        }
        #pragma unroll
        for (int e = 0; e < 8; ++e) {
            const int q = qvals[(krow + 16 + h * 8 + e) * Ndim + n];
            aw[8 + e] = (_Float16)__builtin_fmaf((float)q, sc, nsc8);
        }
        unsigned ipack = 0;
        const int gg0 = (k0 >> 2) + h * 8;
        #pragma unroll
        for (int j = 0; j < 8; ++j) {
            const int2 ii = *(const int2*)(idx + ((size_t)(gg0 + j) * Ndim + n) * 2);
            ipack |= (unsigned)((ii.x & 3) | ((ii.y & 3) << 2)) << (4 * j);
        }

        __syncthreads();

        #pragma unroll
        for (int mt = 0; mt < 4; ++mt) {
            const _Float16* rp = As + (mt * 16 + ln16) * LDSROW + h * 16;
            v16h blo = *(const v16h*)(rp);
            v16h bhi = *(const v16h*)(rp + 32);
            v32h bf  = __builtin_shufflevector(blo, bhi,
                0, 1, 2, 3, 4, 5, 6, 7, 8, 9, 10, 11, 12, 13, 14, 15,
                16, 17, 18, 19, 20, 21, 22, 23, 24, 25, 26, 27, 28, 29, 30, 31);
            acc[mt] = __builtin_amdgcn_swmmac_f32_16x16x64_f16(
                false, aw, false, bf, acc[mt], (int)ipack, false, false);
        }
        __syncthreads();
    }

    float* Tw = (float*)smem + wave * (16 * 17);
    #pragma unroll
    for (int mt = 0; mt < 4; ++mt) {
        #pragma unroll
        for (int r = 0; r < 8; ++r)
            Tw[(r + 8 * h) * 17 + ln16] = acc[mt][r];
        #pragma unroll
        for (int r = 0; r < 8; ++r)
            out[(size_t)(m0 + mt * 16 + r + 8 * h) * Ndim + n] = Tw[ln16 * 17 + (r + 8 * h)];
    }
}

extern "C" void kernel_launch(void* const* d_in, const int* in_sizes, int n_in,
                              void* d_out, int out_size, void* d_ws, size_t ws_size,
                              hipStream_t stream)
{
    const float* A  = (const float*)d_in[0];
    const int*   qv = (const int*)  d_in[1];
    const int*   ix = (const int*)  d_in[2];
    const float* sc = (const float*)d_in[3];
    float*       out = (float*)d_out;

    // grid.x = m-tiles (fast dim) so all M-tiles of one n-range are co-resident:
    // the 256 MB qvals/idx stream is read from HBM once and re-read from L2.
    dim3 grid(Mdim / MT, Ndim / NT);   // (4, 64)

    const size_t needA2 = (size_t)Mdim * Kdim * sizeof(_Float16);   // 4 MB
    if (ws_size >= needA2) {
        _Float16* A2 = (_Float16*)d_ws;
        convA<<<dim3((Mdim * Kdim) / (256 * 8)), dim3(256), 0, stream>>>(A, A2);
        kern_async<<<grid, dim3(256), 0, stream>>>(A2, qv, ix, sc, out);
    } else {
        kern_sync<<<grid, dim3(256), 0, stream>>>(A, qv, ix, sc, out);
    }
}